// PostProcess_66322884985406
// MI455X (gfx1250) — compile-verified
//
#include <hip/hip_runtime.h>
#include <hip/hip_bf16.h>
#include <math.h>

// ---------------------------------------------------------------------------
// Problem constants
// ---------------------------------------------------------------------------
#define BATCH   32
#define NQ      300
#define NC      91
#define NA      10
#define NKEEP   256
#define MLEN    128
#define NTOPK   50
#define PROBN   (NQ * NC)          // 27300

// d_out layout (floats, concatenated in reference return order)
#define OFF_SCORES        0
#define OFF_LABELS        (OFF_SCORES + BATCH * NTOPK)                  // 1600
#define OFF_BOXES         (OFF_LABELS + BATCH * NTOPK)                  // 3200
#define OFF_MASKS         (OFF_BOXES + BATCH * NTOPK * 4)               // 9600
#define OFF_SCORES_INT    (OFF_MASKS + (size_t)BATCH * NTOPK * MLEN * MLEN)
#define OFF_LABELS_INT    (OFF_SCORES_INT + BATCH * NTOPK)
#define OFF_LABELS_ACT    (OFF_LABELS_INT + BATCH * NTOPK)

typedef float v2f __attribute__((ext_vector_type(2)));
typedef float v8f __attribute__((ext_vector_type(8)));

__device__ __forceinline__ float sigmoidf(float x) {
    return 1.0f / (1.0f + expf(-x));
}

// ---------------------------------------------------------------------------
// Kernel A: per-batch sigmoid + iterative top-50 (probs resident in LDS)
// grid = BATCH blocks of 256 threads; dynamic LDS = (PROBN + 512) floats
// ---------------------------------------------------------------------------
__global__ void topk_kernel(const float* __restrict__ logits,
                            float* __restrict__ out,     // d_out base
                            int*   __restrict__ ws_idx) {
    extern __shared__ float sm[];
    float* probs = sm;                 // PROBN
    float* redv  = sm + PROBN;         // 256
    int*   redii = (int*)(redv + 256); // 256

    const int tid = threadIdx.x;
    const int b   = blockIdx.x;
    const float* lg = logits + (size_t)b * PROBN;

    for (int i = tid; i < PROBN; i += 256)
        probs[i] = sigmoidf(lg[i]);
    __syncthreads();

    for (int t = 0; t < NTOPK; ++t) {
        float bv = -1.0f;
        int   bi = 0x7fffffff;
        for (int i = tid; i < PROBN; i += 256) {
            float v = probs[i];
            if (v > bv || (v == bv && i < bi)) { bv = v; bi = i; }
        }
        redv[tid]  = bv;
        redii[tid] = bi;
        __syncthreads();
        for (int s = 128; s > 0; s >>= 1) {
            if (tid < s) {
                float ov = redv[tid + s];
                int   oi = redii[tid + s];
                if (ov > redv[tid] || (ov == redv[tid] && oi < redii[tid])) {
                    redv[tid] = ov; redii[tid] = oi;
                }
            }
            __syncthreads();
        }
        if (tid == 0) {
            int idx = redii[0];
            out[OFF_SCORES + b * NTOPK + t] = redv[0];
            out[OFF_LABELS + b * NTOPK + t] = (float)(idx % NC);
            ws_idx[b * NTOPK + t] = idx;
            probs[idx] = -1.0f;        // exclude for next round
        }
        __syncthreads();
    }
}

// ---------------------------------------------------------------------------
// Kernel B: boxes / interms / actions / vector gather
// ---------------------------------------------------------------------------
__global__ void gather_kernel(const float* __restrict__ bbox,
                              const float* __restrict__ vector,
                              const float* __restrict__ interms,
                              const float* __restrict__ actions,
                              const int*   __restrict__ tsizes,
                              const int*   __restrict__ ws_idx,
                              float* __restrict__ ws_vec,
                              float* __restrict__ out) {
    const int gid = blockIdx.x * blockDim.x + threadIdx.x;
    if (gid < BATCH * NTOPK) {
        const int b = gid / NTOPK;
        const int idx = ws_idx[gid];
        const int q = idx / NC;            // topk_boxes

        // boxes: cxcywh -> xyxy, scaled by (w,h,w,h)
        const float* bb = bbox + ((size_t)b * NQ + q) * 4;
        float cx = bb[0], cy = bb[1], w = bb[2], h = bb[3];
        float img_h = (float)tsizes[b * 2 + 0];
        float img_w = (float)tsizes[b * 2 + 1];
        float* ob = out + OFF_BOXES + (size_t)gid * 4;
        ob[0] = (cx - 0.5f * w) * img_w;
        ob[1] = (cy - 0.5f * h) * img_h;
        ob[2] = (cx + 0.5f * w) * img_w;
        ob[3] = (cy + 0.5f * h) * img_h;

        // interms: sigmoid monotonic -> argmax over logits, sigmoid of max
        const float* it = interms + ((size_t)b * NQ + q) * NC;
        float bv = it[0]; int bi = 0;
        for (int c = 1; c < NC; ++c) {
            float v = it[c];
            if (v > bv) { bv = v; bi = c; }
        }
        out[OFF_SCORES_INT + gid] = sigmoidf(bv);
        out[OFF_LABELS_INT + gid] = (float)bi;

        // gather vectors (16B vectorized; both sides 16B aligned)
        const float4* vv  = (const float4*)(vector + ((size_t)b * NQ + q) * NKEEP);
        float4*       dst = (float4*)(ws_vec + (size_t)gid * NKEEP);
        for (int k = 0; k < NKEEP / 4; ++k) dst[k] = vv[k];
    }
    if (gid < BATCH) {
        // softmax monotonic -> argmax over logits
        const float* ac = actions + (size_t)gid * NA;
        float bv = ac[0]; int bi = 0;
        for (int a = 1; a < NA; ++a) {
            float v = ac[a];
            if (v > bv) { bv = v; bi = a; }
        }
        out[OFF_LABELS_ACT + gid] = (float)bi;
    }
}

// ---------------------------------------------------------------------------
// Kernel C: inverse-zigzag scatter + re_mask = D^T c D via V_WMMA_F32_16X16X4_F32
// K-major (transposed) LDS operand layouts so every A/B pair is one b64 load.
// one workgroup (8 wave32) per (b,q); M kept in LDS for min/max + threshold.
// ---------------------------------------------------------------------------
#define KDIM   24            // padded 23 -> 24 (6 WMMA K-steps of 4)
#define KS     26            // K-major row stride (even -> 8B-aligned pairs)
#define MMS    132           // M row stride (multiple of 4 -> 16B-aligned quads)

__device__ __forceinline__ v8f wmma_f32_4(v2f a, v2f b, v8f c) {
    return __builtin_amdgcn_wmma_f32_16x16x4_f32(false, a, false, b,
                                                 (short)0, c, false, false);
}

__device__ __forceinline__ v2f lds_load2(const float* p) {
    return *(const v2f*)p;              // 8B-aligned by construction
}

__global__ void dct_mask_kernel(const float* __restrict__ ws_vec,
                                float* __restrict__ masks_out) {
    extern __shared__ float sm[];
    float* DtopT = sm;                       // 128 * KS  (n-major, k contig)  3328
    float* cmatT = DtopT + 128 * KS;         //  32 * KS  (l-major, k contig)   832
    float* U     = cmatT + 32 * KS;          // 128 * KS  (n-major, l contig)  3328
    float* M     = U + 128 * KS;             // 128 * MMS                     16896
    float* rmin  = M + 128 * MMS;            // 256
    float* rmax  = rmin + 256;               // 256

    const int tid  = threadIdx.x;
    const int lane = tid & 31;
    const int wv   = tid >> 5;               // wave id 0..7
    const int half = lane >> 4;              // 0 or 1 (K-pair select)
    const int l16  = lane & 15;
    const int bq   = blockIdx.x;             // 0..1599

    // --- DtopT[n][k] = D[k][n] for k<23, zero-padded k=23..25 ---
    for (int i = tid; i < 128 * KS; i += 256) {
        int n = i / KS, k = i - n * KS;
        float v = 0.0f;
        if (k < 23) {
            float scale = (k == 0) ? 0.08838834764831845f : 0.125f;  // sqrt(1/128), sqrt(2/128)
            v = cosf(3.14159265358979323846f * (2.0f * n + 1.0f) * (float)k / 256.0f) * scale;
        }
        DtopT[i] = v;
    }

    // --- cmatT[l][k] = c[k][l]: inverse-zigzag scatter of 256 coefficients ---
    const float* vec = ws_vec + (size_t)bq * NKEEP;
    for (int i = tid; i < 32 * KS; i += 256) {
        int l = i / KS, k = i - l * KS;
        float v = 0.0f;
        if (k < 23 && l < 23) {
            int s = k + l;
            int r = (s * (s + 1)) / 2 + ((s & 1) ? k : (s - k));
            if (r < NKEEP) v = vec[r];
        }
        cmatT[i] = v;
    }
    __syncthreads();

    // --- step 1:  U[n][l] = sum_k D[k][n] * c[k][l]   (128 x 24) ---
    // wave wv owns n-tile wv; l-tiles 0,1.  K-outer: hoist A, 2 live accs.
    {
        v8f acc0 = {}, acc1 = {};
        const float* arow = &DtopT[(16 * wv + l16) * KS];
        const float* brow0 = &cmatT[(l16) * KS];
        const float* brow1 = &cmatT[(16 + l16) * KS];
        for (int kk = 0; kk < 6; ++kk) {
            int kf = 4 * kk + 2 * half;
            v2f a  = lds_load2(arow + kf);     // A[M=n][K=k] = D[k][n]
            v2f b0 = lds_load2(brow0 + kf);    // B[K=k][N=l]
            v2f b1 = lds_load2(brow1 + kf);
            acc0 = wmma_f32_4(a, b0, acc0);
            acc1 = wmma_f32_4(a, b1, acc1);
        }
        for (int j = 0; j < 8; ++j) {
            int row = 16 * wv + j + 8 * half;
            U[row * KS + l16]      = acc0[j];
            U[row * KS + 16 + l16] = acc1[j];
        }
    }
    __syncthreads();

    // --- step 2:  M[n][m] = sum_l U[n][l] * D[l][m]   (128 x 128) ---
    // K-outer: hoist A; 8 independent accumulators -> deep WMMA pipelining.
    float lmin =  3.4e38f, lmax = -3.4e38f;
    {
        v8f acc[8];
        for (int t = 0; t < 8; ++t) acc[t] = (v8f){};
        const float* arow = &U[(16 * wv + l16) * KS];
        for (int kk = 0; kk < 6; ++kk) {
            int kf = 4 * kk + 2 * half;
            v2f a = lds_load2(arow + kf);                        // A[M=n][K=l]
            #pragma unroll
            for (int tm = 0; tm < 8; ++tm) {
                v2f b = lds_load2(&DtopT[(16 * tm + l16) * KS + kf]);  // B[K=l][N=m]
                acc[tm] = wmma_f32_4(a, b, acc[tm]);
            }
        }
        #pragma unroll
        for (int tm = 0; tm < 8; ++tm) {
            #pragma unroll
            for (int j = 0; j < 8; ++j) {
                float v = acc[tm][j];
                M[(16 * wv + j + 8 * half) * MMS + 16 * tm + l16] = v;
                lmin = fminf(lmin, v);
                lmax = fmaxf(lmax, v);
            }
        }
    }
    rmin[tid] = lmin;
    rmax[tid] = lmax;
    __syncthreads();
    for (int s = 128; s > 0; s >>= 1) {
        if (tid < s) {
            rmin[tid] = fminf(rmin[tid], rmin[tid + s]);
            rmax[tid] = fmaxf(rmax[tid], rmax[tid + s]);
        }
        __syncthreads();
    }
    const float thr = 0.5f * (rmin[0] + rmax[0]);

    // threshold + vectorized 16B stores (global_store_b128)
    float* outp = masks_out + (size_t)bq * (MLEN * MLEN);
    for (int base = 0; base < MLEN * MLEN; base += 1024) {
        int i = base + tid * 4;
        int r = i >> 7, c = i & 127;
        float4 m4 = *(const float4*)&M[r * MMS + c];
        float4 o;
        o.x = (m4.x > thr) ? 1.0f : 0.0f;
        o.y = (m4.y > thr) ? 1.0f : 0.0f;
        o.z = (m4.z > thr) ? 1.0f : 0.0f;
        o.w = (m4.w > thr) ? 1.0f : 0.0f;
        *(float4*)&outp[i] = o;
    }
}

// ---------------------------------------------------------------------------
// launch
// ---------------------------------------------------------------------------
extern "C" void kernel_launch(void* const* d_in, const int* in_sizes, int n_in,
                              void* d_out, int out_size, void* d_ws, size_t ws_size,
                              hipStream_t stream) {
    const float* logits  = (const float*)d_in[0];
    const float* bbox    = (const float*)d_in[1];
    const float* vector  = (const float*)d_in[2];
    const float* interms = (const float*)d_in[3];
    const float* actions = (const float*)d_in[4];
    const int*   tsizes  = (const int*)d_in[5];

    float* out    = (float*)d_out;
    int*   ws_idx = (int*)d_ws;
    float* ws_vec = (float*)d_ws + BATCH * NTOPK;   // after 1600 ints (16B-aligned)

    // A: top-k per batch
    size_t shA = (size_t)(PROBN + 512) * sizeof(float);
    topk_kernel<<<BATCH, 256, shA, stream>>>(logits, out, ws_idx);

    // B: gathers + small outputs
    gather_kernel<<<(BATCH * NTOPK + 255) / 256, 256, 0, stream>>>(
        bbox, vector, interms, actions, tsizes, ws_idx, ws_vec, out);

    // C: WMMA DCT reconstruction + threshold
    size_t shC = (size_t)(128 * KS + 32 * KS + 128 * KS + 128 * MMS + 512) * sizeof(float);
    dct_mask_kernel<<<BATCH * NTOPK, 256, shC, stream>>>(ws_vec, out + OFF_MASKS);
}